// _ShotRNN_90417651516362
// MI455X (gfx1250) — compile-verified
//
#include <hip/hip_runtime.h>
#include <math.h>

typedef __attribute__((ext_vector_type(2))) float v2f;
typedef __attribute__((ext_vector_type(8))) float v8f;

#define B_TOT   32768
#define T_STEPS 60
#define FIN     26
#define H       24
// gates padded to 32 columns each -> 96 columns = 6 N-tiles of 16
#define NT      6
#define KCX     7   // ceil(26/4)
#define KCH     6   // 24/4

__device__ __forceinline__ float fast_sig(float x) {
    return __builtin_amdgcn_rcpf(1.0f + __expf(-x));
}
__device__ __forceinline__ float fast_tanh(float x) {
    const float e = __expf(-2.0f * x);
    return (1.0f - e) * __builtin_amdgcn_rcpf(1.0f + e);
}

__global__ __launch_bounds__(128)
void gru_fused_wmma(const float* __restrict__ x,
                    const float* __restrict__ w_ih, const float* __restrict__ w_hh,
                    const float* __restrict__ b_ih, const float* __restrict__ b_hh,
                    const float* __restrict__ fc1_w, const float* __restrict__ fc1_b,
                    const float* __restrict__ fc2_w, const float* __restrict__ fc2_b,
                    float* __restrict__ out)
{
    // B-matrix fragments prepacked in per-lane fragment order (shared by all 4 waves):
    // wf*[ (kc*NT + nt)*32 + lane ] = { W^T[K=kc*4+2*lane_hi+0][N'], W^T[K..+1][N'] },
    // N' = (nt>>1)*32-gate-padded column, valid iff j = ((nt&1)<<4)+lane_lo < 24.
    __shared__ v2f  wfx[KCX * NT * 32];     // 10.5 KB
    __shared__ v2f  wfh[KCH * NT * 32];     //  9.0 KB
    __shared__ float h_lds[4][16 * H];      //  6.0 KB  per-wave hidden tile [m][k]

    const int wave    = threadIdx.x >> 5;
    const int lane    = threadIdx.x & 31;
    const int lane_lo = lane & 15;
    const int lane_hi = lane >> 4;
    const int tile    = blockIdx.x * 4 + wave;
    const int row0    = tile * 16;

    // ---------------- wave 0: pack weight fragments into LDS ---------------
    if (wave == 0) {
        #pragma unroll
        for (int nt = 0; nt < NT; ++nt) {
            const int gg = nt >> 1;
            const int j  = ((nt & 1) << 4) + lane_lo;
            const bool jv = (j < H);
            #pragma unroll
            for (int kc = 0; kc < KCX; ++kc) {
                v2f w;
                #pragma unroll
                for (int v = 0; v < 2; ++v) {
                    const int K = kc * 4 + 2 * lane_hi + v;
                    w[v] = (jv && K < FIN) ? w_ih[(gg * H + j) * FIN + K] : 0.0f;
                }
                wfx[(kc * NT + nt) * 32 + lane] = w;
            }
            #pragma unroll
            for (int kc = 0; kc < KCH; ++kc) {
                v2f w;
                #pragma unroll
                for (int v = 0; v < 2; ++v) {
                    const int K = kc * 4 + 2 * lane_hi + v;   // always < 24
                    w[v] = jv ? w_hh[(gg * H + j) * H + K] : 0.0f;
                }
                wfh[(kc * NT + nt) * 32 + lane] = w;
            }
        }
    }

    // per-lane scalar biases (applied in pointwise, not in accumulator init)
    float bRZ[4], bXN[2], bHN[2];
    #pragma unroll
    for (int nt = 0; nt < 4; ++nt) {
        const int gg = nt >> 1;
        const int j  = ((nt & 1) << 4) + lane_lo;
        bRZ[nt] = (j < H) ? (b_ih[gg * H + j] + b_hh[gg * H + j]) : 0.0f;
    }
    #pragma unroll
    for (int tt = 0; tt < 2; ++tt) {
        const int j = (tt << 4) + lane_lo;
        bXN[tt] = (j < H) ? b_ih[2 * H + j] : 0.0f;
        bHN[tt] = (j < H) ? b_hh[2 * H + j] : 0.0f;
    }

    // h = 0 : registers (C layout) + LDS copy (A-fragment source)
    v8f hreg[2] = {};
    #pragma unroll
    for (int tt = 0; tt < 2; ++tt) {
        const int j = (tt << 4) + lane_lo;
        if (j < H) {
            #pragma unroll
            for (int i = 0; i < 8; ++i)
                h_lds[wave][(i + 8 * lane_hi) * H + j] = 0.0f;
        }
    }
    const v8f zeroC = {};                    // single shared C operand for chain starts
    __syncthreads();

    const float* xrow = x + (size_t)(row0 + lane_lo) * (T_STEPS * FIN);
    const v2f*   wxl  = wfx + lane;          // lane-indexed fragment tables
    const v2f*   whl  = wfh + lane;

    for (int t = 0; t < T_STEPS; ++t) {
        // ---- A fragments: x_t (global, 8B-aligned pairs), h (LDS b64) ----
        v2f ax[KCX];
        const float* xt = xrow + t * FIN;
        #pragma unroll
        for (int kc = 0; kc < KCX; ++kc) {
            const int k0 = kc * 4 + 2 * lane_hi;
            if (k0 < FIN) ax[kc] = *(const v2f*)(xt + k0);
            else          ax[kc] = (v2f){0.0f, 0.0f};
        }
        v2f ah[KCH];
        const float* hl = &h_lds[wave][lane_lo * H + 2 * lane_hi];
        #pragma unroll
        for (int kc = 0; kc < KCH; ++kc)
            ah[kc] = *(const v2f*)(hl + kc * 4);

        v8f accRZ[4], accXN[2], accHN[2];

        // ---- x @ w_ih^T : 7 K-chunks x 6 N-tiles (B frags from LDS) ------
        #pragma unroll
        for (int kc = 0; kc < KCX; ++kc) {
            const v2f a = ax[kc];
            #pragma unroll
            for (int nt = 0; nt < 4; ++nt) {
                const v2f w = wxl[(kc * NT + nt) * 32];
                accRZ[nt] = __builtin_amdgcn_wmma_f32_16x16x4_f32(
                    false, a, false, w, (short)0, (kc == 0) ? zeroC : accRZ[nt], false, false);
            }
            #pragma unroll
            for (int tt = 0; tt < 2; ++tt) {
                const v2f w = wxl[(kc * NT + 4 + tt) * 32];
                accXN[tt] = __builtin_amdgcn_wmma_f32_16x16x4_f32(
                    false, a, false, w, (short)0, (kc == 0) ? zeroC : accXN[tt], false, false);
            }
        }
        // ---- h @ w_hh^T : 6 K-chunks x 6 N-tiles -------------------------
        #pragma unroll
        for (int kc = 0; kc < KCH; ++kc) {
            const v2f a = ah[kc];
            #pragma unroll
            for (int nt = 0; nt < 4; ++nt) {
                const v2f w = whl[(kc * NT + nt) * 32];
                accRZ[nt] = __builtin_amdgcn_wmma_f32_16x16x4_f32(
                    false, a, false, w, (short)0, accRZ[nt], false, false);
            }
            #pragma unroll
            for (int tt = 0; tt < 2; ++tt) {
                const v2f w = whl[(kc * NT + 4 + tt) * 32];
                accHN[tt] = __builtin_amdgcn_wmma_f32_16x16x4_f32(
                    false, a, false, w, (short)0, (kc == 0) ? zeroC : accHN[tt], false, false);
            }
        }

        // ---- gates (lane-local: r_j, z_j, n_j share a lane) --------------
        #pragma unroll
        for (int tt = 0; tt < 2; ++tt) {
            const int j = (tt << 4) + lane_lo;
            #pragma unroll
            for (int i = 0; i < 8; ++i) {
                const float r  = fast_sig(accRZ[tt][i] + bRZ[tt]);
                const float z  = fast_sig(accRZ[2 + tt][i] + bRZ[2 + tt]);
                const float n  = fast_tanh(accXN[tt][i] + bXN[tt] + r * (accHN[tt][i] + bHN[tt]));
                const float hn = (1.0f - z) * n + z * hreg[tt][i];
                hreg[tt][i] = hn;
                if (j < H) h_lds[wave][(i + 8 * lane_hi) * H + j] = hn;
            }
        }
        __syncthreads();   // cross-lane h stores -> next iteration's A-frag loads
    }

    // ---------------- FC head: 24 -> 8 (relu) -> 4 -> softmax --------------
    if (lane < 16) {
        const float* hrow = &h_lds[wave][lane * H];
        float h1[8];
        #pragma unroll
        for (int o = 0; o < 8; ++o) {
            float a = fc1_b[o];
            #pragma unroll
            for (int j = 0; j < H; ++j) a += hrow[j] * fc1_w[o * H + j];
            h1[o] = fmaxf(a, 0.0f);
        }
        float lg[4]; float mx = -3.0e38f;
        #pragma unroll
        for (int c = 0; c < 4; ++c) {
            float a = fc2_b[c];
            #pragma unroll
            for (int o = 0; o < 8; ++o) a += h1[o] * fc2_w[c * 8 + o];
            lg[c] = a; mx = fmaxf(mx, a);
        }
        float s = 0.0f;
        #pragma unroll
        for (int c = 0; c < 4; ++c) { lg[c] = __expf(lg[c] - mx); s += lg[c]; }
        const float inv = 1.0f / s;
        float4 o4 = make_float4(lg[0] * inv, lg[1] * inv, lg[2] * inv, lg[3] * inv);
        *(float4*)(out + (size_t)(row0 + lane) * 4) = o4;
    }
}

extern "C" void kernel_launch(void* const* d_in, const int* in_sizes, int n_in,
                              void* d_out, int out_size, void* d_ws, size_t ws_size,
                              hipStream_t stream) {
    (void)in_sizes; (void)n_in; (void)d_ws; (void)ws_size; (void)out_size;
    const float* x     = (const float*)d_in[0];
    const float* w_ih  = (const float*)d_in[1];
    const float* w_hh  = (const float*)d_in[2];
    const float* b_ih  = (const float*)d_in[3];
    const float* b_hh  = (const float*)d_in[4];
    const float* fc1_w = (const float*)d_in[5];
    const float* fc1_b = (const float*)d_in[6];
    const float* fc2_w = (const float*)d_in[7];
    const float* fc2_b = (const float*)d_in[8];

    const int tiles  = B_TOT / 16;        // 2048 waves
    const int blocks = tiles / 4;         // 4 waves (128 threads) per block
    gru_fused_wmma<<<blocks, 128, 0, stream>>>(
        x, w_ih, w_hh, b_ih, b_hh, fc1_w, fc1_b, fc2_w, fc2_b, (float*)d_out);
}